// GPSDepthAttentionLayer_2147483648107
// MI455X (gfx1250) — compile-verified
//
#include <hip/hip_runtime.h>

typedef float v2f __attribute__((ext_vector_type(2)));
typedef float v8f __attribute__((ext_vector_type(8)));

#define OUTF 256
#define HID 16
#define BN_EPS 1e-5f
#define ALPHA 0.2f

// ============================================================================
// WMMA fp32 GEMM:  C[M x Nc] = A[M x 256] * W[256 x Nc] + bias[Nc]
// One wave32 computes one 16x16 output tile using V_WMMA_F32_16X16X4_F32.
// A frag (16x4): lane L -> M = L%16, K = vgpr + 2*(L/16)   (contiguous pair)
// B frag (4x16): lane L -> N = L%16, K = vgpr + 2*(L/16)
// C/D (16x16):   lane L -> N = L%16, M = vgpr + 8*(L/16)
// ============================================================================
__global__ void wmma_gemm_k256(const float* __restrict__ A,
                               const float* __restrict__ Bw,
                               const float* __restrict__ bias,
                               float* __restrict__ Cout,
                               int M, int Nc) {
  int waveId = (int)((blockIdx.x * blockDim.x + threadIdx.x) >> 5);
  int lane   = threadIdx.x & 31;
  int ntn    = Nc >> 4;
  int total  = (M >> 4) * ntn;
  if (waveId >= total) return;              // wave-uniform: EXEC stays all-ones
  int mt = waveId / ntn;
  int nt = waveId - mt * ntn;
  int m0 = mt << 4, n0 = nt << 4;
  int r = lane & 15, half = lane >> 4;

  const float* arow = A  + (size_t)(m0 + r) * 256 + 2 * half;
  const float* bcol = Bw + (size_t)(2 * half) * Nc + (n0 + r);

  v8f acc = {};
  #pragma unroll 8
  for (int k = 0; k < 256; k += 4) {
    v2f a = *(const v2f*)(arow + k);        // A[m0+r][k+2h], A[m0+r][k+2h+1]
    v2f b;
    b.x = bcol[(size_t)k * Nc];             // W[k+2h  ][n0+r]
    b.y = bcol[(size_t)(k + 1) * Nc];       // W[k+2h+1][n0+r]
    acc = __builtin_amdgcn_wmma_f32_16x16x4_f32(false, a, false, b,
                                                (short)0, acc, false, false);
  }

  float bv = bias[n0 + r];
  float* crow = Cout + (size_t)(m0 + 8 * half) * Nc + (n0 + r);
  #pragma unroll
  for (int p = 0; p < 8; ++p) crow[(size_t)p * Nc] = acc[p] + bv;
}

// ============================================================================
// Per-node attention scores: att_la = new_h @ la + Bla ; att_ra likewise.
// One wave per node; 8 cols per lane; wave32 xor-shuffle reduction.
// ============================================================================
__global__ void att_scores(const float* __restrict__ new_h,
                           const float* __restrict__ la,
                           const float* __restrict__ ra,
                           const float* __restrict__ bla,
                           const float* __restrict__ bra,
                           float* __restrict__ att_la,
                           float* __restrict__ att_ra, int Nn) {
  int wid  = (int)((blockIdx.x * blockDim.x + threadIdx.x) >> 5);
  int lane = threadIdx.x & 31;
  if (wid >= Nn) return;
  const float* row = new_h + (size_t)wid * OUTF;
  float pl = 0.f, pr = 0.f;
  #pragma unroll
  for (int j = 0; j < 8; ++j) {
    int c = lane + 32 * j;
    float v = row[c];
    pl += v * la[c];
    pr += v * ra[c];
  }
  #pragma unroll
  for (int off = 16; off > 0; off >>= 1) {
    pl += __shfl_xor(pl, off, 32);
    pr += __shfl_xor(pr, off, 32);
  }
  if (lane == 0) {
    att_la[wid] = pl + bla[0];
    att_ra[wid] = pr + bra[0];
  }
}

// ============================================================================
// Per-edge attention weight + row_sum + edge count (segment sums via atomics)
// ============================================================================
__global__ void edge_att(const int* __restrict__ src, const int* __restrict__ dst,
                         const float* __restrict__ att_la, const float* __restrict__ att_ra,
                         float* __restrict__ a_edge, float* __restrict__ row_sum,
                         float* __restrict__ cnt, int Ne) {
  int e = blockIdx.x * blockDim.x + threadIdx.x;
  if (e >= Ne) return;
  int s = src[e], d = dst[e];
  float a = (att_la[s] + att_ra[d]) * 0.0625f;      // / sqrt(256)
  a = (a >= 0.f) ? a : ALPHA * a;                   // leaky relu
  a = __expf(-a);
  a_edge[e] = a;
  atomicAdd(&row_sum[s], a);
  atomicAdd(&cnt[s], 1.0f);
}

// ============================================================================
// agg[src] += a_edge * new_h[dst]   — one wave per edge, 8 f32 atomics/lane.
// ============================================================================
__global__ void edge_aggregate(const int* __restrict__ src, const int* __restrict__ dst,
                               const float* __restrict__ a_edge,
                               const float* __restrict__ new_h,
                               float* __restrict__ agg, int Ne) {
  int wid  = (int)((blockIdx.x * blockDim.x + threadIdx.x) >> 5);
  int lane = threadIdx.x & 31;
  if (wid >= Ne) return;
  int s = src[wid], d = dst[wid];
  float a = a_edge[wid];
  const float* hr = new_h + (size_t)d * OUTF;
  float* ar = agg + (size_t)s * OUTF;
  #pragma unroll
  for (int j = 0; j < 8; ++j) {
    int c = lane + 32 * j;
    atomicAdd(&ar[c], a * hr[c]);
  }
}

// ============================================================================
// final_h (pre-BN) in place into agg; accumulate per-column sum / sumsq.
// One thread = one column, 128 rows per block.
// ============================================================================
__global__ void final_bn_stats(const float* __restrict__ new_h,
                               const float* __restrict__ row_sum,
                               const float* __restrict__ af_p,
                               float* __restrict__ aggio,
                               float* __restrict__ colsum,
                               float* __restrict__ colsq, int Nn) {
  int c  = threadIdx.x;                 // 0..255
  int r0 = blockIdx.x * 128;
  float af = af_p[0];
  float omaf = 1.f - af;
  float ls = 0.f, lss = 0.f;
  for (int i = 0; i < 128; ++i) {
    int rr = r0 + i;
    if (rr >= Nn) break;
    float rs = row_sum[rr];
    rs = (rs > 0.f) ? rs : 1.f;
    size_t idx = (size_t)rr * OUTF + c;
    float f = af * (aggio[idx] / rs) + omaf * new_h[idx];
    aggio[idx] = f;
    ls += f;
    lss += f * f;
  }
  atomicAdd(&colsum[c], ls);
  atomicAdd(&colsq[c], lss);
}

__global__ void bn_params(const float* __restrict__ colsum, const float* __restrict__ colsq,
                          float* __restrict__ meanv, float* __restrict__ invv, int Nn) {
  int c = threadIdx.x;
  if (c >= OUTF) return;
  float invN = 1.f / (float)Nn;
  float mean = colsum[c] * invN;
  float var  = colsq[c] * invN - mean * mean;
  meanv[c] = mean;
  invv[c]  = rsqrtf(var + BN_EPS);
}

// normalize + relu -> final_h output region
__global__ void bn_norm_relu(const float* __restrict__ aggio,
                             const float* __restrict__ meanv,
                             const float* __restrict__ invv,
                             float* __restrict__ outp, size_t total) {
  size_t i = (size_t)blockIdx.x * blockDim.x + threadIdx.x;
  if (i >= total) return;
  int c = (int)(i & (OUTF - 1));
  float v = (aggio[i] - meanv[c]) * invv[c];
  outp[i] = (v > 0.f) ? v : 0.f;
}

// ============================================================================
// Per-edge MLP: fc = sigmoid(lf2( tanh(lf1([hs, hd, |hd-hs|])) ))
// Weights staged in LDS. One thread per edge.
// ============================================================================
__global__ void edge_mlp(const int* __restrict__ src, const int* __restrict__ dst,
                         const float* __restrict__ mini,
                         const float* __restrict__ w1, const float* __restrict__ b1,
                         const float* __restrict__ w2, const float* __restrict__ b2,
                         float* __restrict__ fc_out, float* __restrict__ sum_fc, int Ne) {
  __shared__ float s_w1[HID * HID * 3];   // 16 x 48
  __shared__ float s_b1[HID];
  __shared__ float s_w2[HID];
  __shared__ float s_b2;
  for (int i = threadIdx.x; i < HID * HID * 3; i += blockDim.x) s_w1[i] = w1[i];
  if (threadIdx.x < HID) { s_b1[threadIdx.x] = b1[threadIdx.x]; s_w2[threadIdx.x] = w2[threadIdx.x]; }
  if (threadIdx.x == 0) s_b2 = b2[0];
  __syncthreads();

  int e = blockIdx.x * blockDim.x + threadIdx.x;
  if (e >= Ne) return;
  int s = src[e], d = dst[e];

  float hs[HID], hd[HID], ad[HID];
  const float4* ps = (const float4*)(mini + (size_t)s * HID);
  const float4* pd = (const float4*)(mini + (size_t)d * HID);
  #pragma unroll
  for (int q = 0; q < 4; ++q) {
    float4 a = ps[q], b = pd[q];
    hs[4*q+0] = a.x; hs[4*q+1] = a.y; hs[4*q+2] = a.z; hs[4*q+3] = a.w;
    hd[4*q+0] = b.x; hd[4*q+1] = b.y; hd[4*q+2] = b.z; hd[4*q+3] = b.w;
  }
  #pragma unroll
  for (int j = 0; j < HID; ++j) ad[j] = fabsf(hd[j] - hs[j]);

  float z = s_b2;
  #pragma unroll
  for (int h = 0; h < HID; ++h) {
    float acc = s_b1[h];
    const float* wr = &s_w1[h * (HID * 3)];
    #pragma unroll
    for (int j = 0; j < HID; ++j) {
      acc += hs[j] * wr[j] + hd[j] * wr[HID + j] + ad[j] * wr[2 * HID + j];
    }
    z += tanhf(acc) * s_w2[h];
  }
  float sg = 1.f / (1.f + __expf(-z));
  fc_out[e] = sg;
  atomicAdd(&sum_fc[s], sg);
}

__global__ void node_div(const float* __restrict__ num, const float* __restrict__ cnt,
                         float* __restrict__ outp, int Nn) {
  int i = blockIdx.x * blockDim.x + threadIdx.x;
  if (i >= Nn) return;
  float c = cnt[i];
  c = (c > 0.f) ? c : 1.f;
  outp[i] = num[i] / c;
}

__global__ void edge_hop2(const int* __restrict__ src, const int* __restrict__ dst,
                          const float* __restrict__ fc, const float* __restrict__ f1,
                          float* __restrict__ sum2, int Ne) {
  int e = blockIdx.x * blockDim.x + threadIdx.x;
  if (e >= Ne) return;
  atomicAdd(&sum2[src[e]], fc[e] * f1[dst[e]]);
}

// ============================================================================
extern "C" void kernel_launch(void* const* d_in, const int* in_sizes, int n_in,
                              void* d_out, int out_size, void* d_ws, size_t ws_size,
                              hipStream_t stream) {
  const float* input  = (const float*)d_in[0];
  const float* af_p   = (const float*)d_in[2];   // aggr_factor
  const int*   edges  = (const int*)d_in[3];     // (2, E) flat
  const float* W      = (const float*)d_in[4];
  const float* Bvec   = (const float*)d_in[5];
  const float* la     = (const float*)d_in[6];
  const float* ra     = (const float*)d_in[7];
  const float* Bla    = (const float*)d_in[8];
  const float* Bra    = (const float*)d_in[9];
  const float* W2     = (const float*)d_in[10];
  const float* B2     = (const float*)d_in[11];
  const float* lf1w   = (const float*)d_in[12];
  const float* lf1b   = (const float*)d_in[13];
  const float* lf2w   = (const float*)d_in[14];
  const float* lf2b   = (const float*)d_in[15];
  (void)n_in; (void)ws_size; (void)out_size;

  const int N_ = in_sizes[0] / OUTF;
  const int E_ = in_sizes[3] / 2;
  const int* src = edges;
  const int* dst = edges + E_;

  float* outp = (float*)d_out;                   // [N*256 final_h][N factor_2hop]

  // -------- workspace layout (floats) --------
  float* w = (float*)d_ws;
  size_t p = 0;
  float* new_h  = w + p; p += (size_t)N_ * OUTF;
  float* att_la = w + p; p += N_;
  float* att_ra = w + p; p += N_;
  float* f1     = w + p; p += N_;
  float* aedge  = w + p; p += E_;                // reused later as fc
  float* mini   = w + p; p += (size_t)N_ * HID;
  float* meanv  = w + p; p += OUTF;
  float* invv   = w + p; p += OUTF;
  // zero-initialized region (accumulators)
  float* zbase  = w + p;
  float* agg    = w + p; p += (size_t)N_ * OUTF;
  float* row_sum= w + p; p += N_;
  float* cnt    = w + p; p += N_;
  float* sum_fc = w + p; p += N_;
  float* sum2   = w + p; p += N_;
  float* colsum = w + p; p += OUTF;
  float* colsq  = w + p; p += OUTF;
  size_t zbytes = ((size_t)N_ * OUTF + 4 * (size_t)N_ + 2 * OUTF) * sizeof(float);
  hipMemsetAsync(zbase, 0, zbytes, stream);

  const int thr = 256;

  // 1) new_h = input @ W + B          (WMMA fp32)
  {
    long long waves = ((long long)(N_ / 16)) * (OUTF / 16);
    int blocks = (int)((waves * 32 + thr - 1) / thr);
    wmma_gemm_k256<<<blocks, thr, 0, stream>>>(input, W, Bvec, new_h, N_, OUTF);
  }
  // 2) attention scores
  {
    int blocks = (N_ + 7) / 8;
    att_scores<<<blocks, thr, 0, stream>>>(new_h, la, ra, Bla, Bra, att_la, att_ra, N_);
  }
  // 3) per-edge attention weight + row_sum + cnt
  {
    int blocks = (E_ + thr - 1) / thr;
    edge_att<<<blocks, thr, 0, stream>>>(src, dst, att_la, att_ra, aedge, row_sum, cnt, E_);
  }
  // 4) agg[src] += a * new_h[dst]
  {
    int blocks = (E_ + 7) / 8;
    edge_aggregate<<<blocks, thr, 0, stream>>>(src, dst, aedge, new_h, agg, E_);
  }
  // 5) final_h (pre-BN) + column stats
  {
    int blocks = (N_ + 127) / 128;
    final_bn_stats<<<blocks, thr, 0, stream>>>(new_h, row_sum, af_p, agg, colsum, colsq, N_);
  }
  // 6) BN mean / inv-std
  bn_params<<<1, OUTF, 0, stream>>>(colsum, colsq, meanv, invv, N_);
  // 7) normalize + relu -> output final_h
  {
    size_t total = (size_t)N_ * OUTF;
    int blocks = (int)((total + thr - 1) / thr);
    bn_norm_relu<<<blocks, thr, 0, stream>>>(agg, meanv, invv, outp, total);
  }
  // 8) mini = final_h @ W_2mini + B_2mini   (WMMA fp32, Nc = 16)
  {
    long long waves = ((long long)(N_ / 16)) * (HID / 16);
    int blocks = (int)((waves * 32 + thr - 1) / thr);
    wmma_gemm_k256<<<blocks, thr, 0, stream>>>(outp, W2, B2, mini, N_, HID);
  }
  // 9) per-edge MLP -> fc (reuses aedge buffer) + segment sum
  {
    int blocks = (E_ + thr - 1) / thr;
    edge_mlp<<<blocks, thr, 0, stream>>>(src, dst, mini, lf1w, lf1b, lf2w, lf2b,
                                         aedge, sum_fc, E_);
  }
  // 10) factor_1hop = sum_fc / cnt
  {
    int blocks = (N_ + thr - 1) / thr;
    node_div<<<blocks, thr, 0, stream>>>(sum_fc, cnt, f1, N_);
  }
  // 11) sum2[src] += fc * f1[dst]
  {
    int blocks = (E_ + thr - 1) / thr;
    edge_hop2<<<blocks, thr, 0, stream>>>(src, dst, aedge, f1, sum2, E_);
  }
  // 12) factor_2hop -> output tail
  {
    int blocks = (N_ + thr - 1) / thr;
    node_div<<<blocks, thr, 0, stream>>>(sum2, cnt, outp + (size_t)N_ * OUTF, N_);
  }
}